// AdaptiveStdPooling2d_82678120448338
// MI455X (gfx1250) — compile-verified
//
#include <hip/hip_runtime.h>

// AdaptiveStdPooling2d: x[16,128,512,128] f32 -> out[16,128,8,16] f32
// HBM-bound (512 MB @ 23.3 TB/s ~= 22 us floor). One block per 64x128 tile.
// Wave-local async global->LDS staging (16 b128 ops in flight per wave,
// per-wave ASYNCcnt wait, no barriers in the hot path), b128 LDS readback,
// single barrier for the cross-wave partial combine, 8-lane shuffle kw-sum.

namespace {
constexpr int KH   = 64;   // rows reduced per bin (variance axis)
constexpr int WW   = 128;  // row width (contiguous)
constexpr int WOUT = 16;   // outputs per tile (WW / kw)
}

typedef int v4i __attribute__((ext_vector_type(4)));
typedef __attribute__((address_space(1))) v4i* g_v4i_ptr;
typedef __attribute__((address_space(3))) v4i* l_v4i_ptr;

#if defined(__has_builtin)
#  if __has_builtin(__builtin_amdgcn_global_load_async_to_lds_b128)
#    define ASYNC_B128_BUILTIN 1
#  endif
#  if __has_builtin(__builtin_amdgcn_s_wait_asynccnt)
#    define WAIT_ASYNC_BUILTIN 1
#  endif
#endif

__device__ __forceinline__ void async_load_b128(const float* g, float* l) {
#if defined(ASYNC_B128_BUILTIN)
  __builtin_amdgcn_global_load_async_to_lds_b128(
      (g_v4i_ptr)(float*)g,   // global src (v4i, AS1)
      (l_v4i_ptr)l,           // LDS dst (v4i, AS3)
      /*offset=*/0, /*cpol=*/0);
#else
  unsigned loff = (unsigned)(unsigned long long)
      (__attribute__((address_space(3))) float*)l;
  asm volatile("global_load_async_to_lds_b128 %0, %1, off"
               :: "v"(loff), "v"(g) : "memory");
#endif
}

__device__ __forceinline__ void wait_async0() {
#if defined(WAIT_ASYNC_BUILTIN)
  __builtin_amdgcn_s_wait_asynccnt(0);
#else
  asm volatile("s_wait_asynccnt 0" ::: "memory");
#endif
  asm volatile("" ::: "memory");  // keep LDS reads below the wait
}

__global__ __launch_bounds__(128)
void AdaptiveStdPooling2d_kernel(const float* __restrict__ x,
                                 float* __restrict__ out) {
  __shared__ __align__(16) float buf[KH * WW];  // 32 KB tile
  __shared__ __align__(16) float p1[4][WW];     // 2 KB partial sums
  __shared__ __align__(16) float p2[4][WW];     // 2 KB partial sum-squares

  const int tile = blockIdx.x;                       // (b*C + c)*H_OUT + ho
  const float* base = x + (size_t)tile * (KH * WW);  // tiles are contiguous
  const int t    = threadIdx.x;   // 0..127
  const int wave = t >> 5;        // row-quarter: rows [16*wave, 16*wave+16)
  const int lane = t & 31;        // byte chunk [16*lane, 16*lane+16) per row

  // Wave-local async staging: wave j loads its own 16 rows (8 KB),
  // 16 b128 async ops in flight per wave, no cross-wave dependency.
  const int off = (wave * 16) * WW + lane * 4;  // floats
  {
    const float* g = base + off;
    float* l = &buf[off];
#pragma unroll
    for (int i = 0; i < 16; ++i)
      async_load_b128(g + i * WW, l + i * WW);
  }

  // Per-wave wait: each lane reads back exactly the chunk this wave's own
  // async instructions wrote, so asynccnt<=0 fully orders it.
  wait_async0();

  // Phase 1 (wave-local, barrier-free): partial sums for columns
  // 4*lane..4*lane+3 over this wave's 16 rows, b128 LDS reads.
  float a0 = 0.f, a1 = 0.f, a2 = 0.f, a3 = 0.f;   // sum x
  float q0 = 0.f, q1 = 0.f, q2 = 0.f, q3 = 0.f;   // sum x^2
  const float4* bp = (const float4*)&buf[off];
#pragma unroll
  for (int i = 0; i < 16; ++i) {
    float4 v = bp[i * (WW / 4)];
    a0 += v.x; q0 = fmaf(v.x, v.x, q0);
    a1 += v.y; q1 = fmaf(v.y, v.y, q1);
    a2 += v.z; q2 = fmaf(v.z, v.z, q2);
    a3 += v.w; q3 = fmaf(v.w, v.w, q3);
  }
  *(float4*)&p1[wave][lane * 4] = make_float4(a0, a1, a2, a3);
  *(float4*)&p2[wave][lane * 4] = make_float4(q0, q1, q2, q3);

  __syncthreads();  // the only barrier: publish partials across waves

  // Combine the 4 row-quarters; thread t owns column t (bank t%64: no conflicts).
  float s1 = (p1[0][t] + p1[1][t]) + (p1[2][t] + p1[3][t]);
  float s2 = (p2[0][t] + p2[1][t]) + (p2[2][t] + p2[3][t]);

  const float inv = 1.0f / (float)KH;
  float mean = s1 * inv;
  float var  = fmaf(-mean, mean, s2 * inv);
  var = fmaxf(var, 0.0f) + 1e-14f;
  float sd = sqrtf(var);

  // Sum stds over kw=8 consecutive columns (8-lane segments of the wave)
  sd += __shfl_xor(sd, 1, 8);
  sd += __shfl_xor(sd, 2, 8);
  sd += __shfl_xor(sd, 4, 8);

  if ((t & 7) == 0)
    out[(size_t)tile * WOUT + (t >> 3)] = sd;
}

extern "C" void kernel_launch(void* const* d_in, const int* in_sizes, int n_in,
                              void* d_out, int out_size, void* d_ws, size_t ws_size,
                              hipStream_t stream) {
  const float* x = (const float*)d_in[0];
  float* out = (float*)d_out;
  const int tiles = in_sizes[0] / (KH * WW);  // 16 * 128 * 8 = 16384
  hipLaunchKernelGGL(AdaptiveStdPooling2d_kernel,
                     dim3(tiles), dim3(128), 0, stream, x, out);
}